// GraphConvNet_50766513439200
// MI455X (gfx1250) — compile-verified
//
#include <hip/hip_runtime.h>
#include <hip/hip_bf16.h>

#define BATCH 8192
#define JNT   17
#define DIN   2048
#define NP    32          // padded N for WMMA (17 -> 2 x 16)
#define VSQ   (JNT * JNT) // 289
#define NBLK  9           // 1 big conv + 8 small convs (BN stat sets)
#define KC    64          // K-chunk staged per TDM transfer
#define LROW  68          // padded LDS row stride in floats (64 + 4 pad dwords)

typedef __attribute__((ext_vector_type(2))) float        v2f;
typedef __attribute__((ext_vector_type(8))) float        v8f;
typedef __attribute__((ext_vector_type(4))) unsigned int v4u;
typedef __attribute__((ext_vector_type(8))) unsigned int v8u;

// adjacency (I + symmetric COCO skeleton) as per-row bitmasks
__device__ const unsigned int ADJ_MASK[JNT] = {
    0x00007u, 0x0000Fu, 0x00017u, 0x0002Au, 0x00054u, 0x008E8u,
    0x01170u, 0x002A0u, 0x00540u, 0x00280u, 0x00500u, 0x03820u,
    0x05840u, 0x0A800u, 0x15000u, 0x0A000u, 0x14000u};

// ---------------------------------------------------------------------------
__global__ void zero_buf(float* __restrict__ p, int n) {
    int i = blockIdx.x * blockDim.x + threadIdx.x;
    if (i < n) p[i] = 0.0f;
}

// WT[v][o][d] = (o < 17) ? W0[v][d][o] : 0   (pad N to 32, make B-frags b64)
__global__ void transpose_w0(const float* __restrict__ W0, float* __restrict__ WT) {
    int idx = blockIdx.x * blockDim.x + threadIdx.x;
    if (idx >= JNT * NP * DIN) return;
    int d = idx & (DIN - 1);
    int o = (idx >> 11) & 31;
    int v = idx >> 16;
    WT[idx] = (o < JNT) ? W0[((size_t)v * DIN + d) * JNT + o] : 0.0f;
}

// ---------------------------------------------------------------------------
// Big GEMM: per vertex v, H[:, v, :] = img_feat[:, v, :] @ W0[v] + b0[v]
// wave32 fp32 WMMA 16x16x4. One 16-row M-tile per wave (8704 waves total).
// A tiles are DMA'd to LDS by the Tensor Data Mover (double-buffered,
// bank-conflict-free via D# padding); B fragments come from L0/L2.
__global__ __launch_bounds__(128) void gemm_big(
    const float* __restrict__ A,     // img_feat [B,17,2048]
    const float* __restrict__ WT,    // [17,32,2048]
    const float* __restrict__ bias,  // b0 [17,17]
    float* __restrict__ H,           // [B,17,17]
    float* __restrict__ sums)        // [2][289] sum, sumsq
{
    __shared__ float lbuf[4][2][16 * LROW];  // 4 waves x double buffer x tile

    const int v    = blockIdx.y;
    const int wave = __builtin_amdgcn_readfirstlane((int)(threadIdx.x >> 5));
    const int lane = threadIdx.x & 31;
    const int half = lane >> 4;
    const int lm   = lane & 15;
    const int wg   = blockIdx.x * 4 + wave;  // 0..511 per vertex
    const int row0 = wg * 16;

    // B fragments: lane<16 holds (K=k,k+1, N=lm); lane>=16 holds (K=k+2,k+3)
    const float* b0p = WT + ((size_t)(v * NP + lm) * DIN) + 2 * half;
    const float* b1p = WT + ((size_t)(v * NP + 16 + lm) * DIN) + 2 * half;
    const float* a0  = A + ((size_t)row0 * JNT + v) * DIN;  // wave-uniform

    const unsigned lds0 = (unsigned)(uintptr_t)&lbuf[wave][0][0];
    const unsigned lds1 = (unsigned)(uintptr_t)&lbuf[wave][1][0];

    // TDM descriptor group1 (chunk-invariant): 4B elems; LDS pad 4 dwords
    // after every 64 dwords (row stride 272B -> ds reads hit banks 4*lm).
    v8u g1;
    g1[0] = (2u << 16) | (1u << 20) | (5u << 22) | (3u << 25);
    g1[1] = (unsigned)KC << 16;          // tensor_dim0 = 64
    g1[2] = 16u << 16;                   // tensor_dim1 = 16
    g1[3] = (unsigned)KC << 16;          // tile_dim0   = 64
    g1[4] = 16u;                         // tile_dim1   = 16
    g1[5] = (unsigned)(JNT * DIN);       // tensor_dim0_stride = 34816
    g1[6] = 0u;
    g1[7] = 0u;

    auto issue_tdm = [&](int chunk) {
        unsigned long long ga = (unsigned long long)(uintptr_t)(a0 + chunk * KC);
        v4u g0;
        g0[0] = 1u;                                   // count=1, no gather
        g0[1] = (chunk & 1) ? lds1 : lds0;            // lds_addr
        g0[2] = (unsigned)ga;                         // global_addr lo
        g0[3] = ((unsigned)(ga >> 32) & 0x01FFFFFFu) | 0x80000000u;  // hi|type=2
        asm volatile("tensor_load_to_lds %0, %1" ::"s"(g0), "s"(g1) : "memory");
    };

    issue_tdm(0);

    const float bias0 = bias[v * JNT + lm];
    const float bias1 = bias[v * JNT + 16];

    v8f acc0 = {};
    v8f acc1 = {};
    for (int c = 0; c < DIN / KC; ++c) {
        if (c + 1 < DIN / KC) {
            issue_tdm(c + 1);
            asm volatile("s_wait_tensorcnt 0x1" ::: "memory");
        } else {
            asm volatile("s_wait_tensorcnt 0x0" ::: "memory");
        }
        const float* la  = &lbuf[wave][c & 1][lm * LROW + 2 * half];
        const float* bb0 = b0p + c * KC;
        const float* bb1 = b1p + c * KC;
#pragma unroll
        for (int k = 0; k < KC; k += 4) {
            v2f af  = *(const v2f*)(la + k);
            v2f bf0 = *(const v2f*)(bb0 + k);
            v2f bf1 = *(const v2f*)(bb1 + k);
            acc0 = __builtin_amdgcn_wmma_f32_16x16x4_f32(
                false, af, false, bf0, (short)0, acc0, false, false);
            acc1 = __builtin_amdgcn_wmma_f32_16x16x4_f32(
                false, af, false, bf1, (short)0, acc1, false, false);
        }
    }

    // C layout: VGPR i -> row = row0 + i + 8*half, col = lm (acc0) / 16 (acc1)
    float s0 = 0.0f, q0 = 0.0f, s1 = 0.0f, q1 = 0.0f;
#pragma unroll
    for (int i = 0; i < 8; ++i) {
        const int r = row0 + i + 8 * half;
        float h0   = acc0[i] + bias0;
        H[(size_t)r * VSQ + v * JNT + lm] = h0;
        s0 += h0;
        q0 += h0 * h0;
        if (lm == 0) {
            float h1 = acc1[i] + bias1;
            H[(size_t)r * VSQ + v * JNT + 16] = h1;
            s1 += h1;
            q1 += h1 * h1;
        }
    }
    atomicAdd(&sums[v * JNT + lm], s0);
    atomicAdd(&sums[VSQ + v * JNT + lm], q0);
    if (lm == 0) {
        atomicAdd(&sums[v * JNT + 16], s1);
        atomicAdd(&sums[VSQ + v * JNT + 16], q1);
    }
}

// ---------------------------------------------------------------------------
__global__ void finalize_stats(const float* __restrict__ sums,
                               const float* __restrict__ g,
                               const float* __restrict__ be,
                               float* __restrict__ stats) {
    int i = blockIdx.x * blockDim.x + threadIdx.x;
    if (i >= VSQ) return;
    const float inv  = 1.0f / (float)BATCH;
    float mean = sums[i] * inv;
    float var  = sums[VSQ + i] * inv - mean * mean;
    float sc   = g[i] * rsqrtf(var + 1e-5f);
    stats[i]       = sc;                 // scale
    stats[VSQ + i] = be[i] - mean * sc;  // shift
}

// BN affine + adjacency aggregation + ReLU (+ optional residual)
__global__ __launch_bounds__(256) void bn_adj_relu(
    const float* __restrict__ H, const float* __restrict__ stats,
    const float* __restrict__ res, float* __restrict__ out) {
    int idx = blockIdx.x * blockDim.x + threadIdx.x;  // b*17 + o
    if (idx >= BATCH * JNT) return;
    const int o = idx % JNT;
    const int b = idx / JNT;

    float hn[JNT];
    const float* hb = H + (size_t)b * VSQ + o;
#pragma unroll
    for (int u = 0; u < JNT; ++u)
        hn[u] = hb[u * JNT] * stats[u * JNT + o] + stats[VSQ + u * JNT + o];

    float* ob       = out + (size_t)b * VSQ + o;
    const float* rb = res ? res + (size_t)b * VSQ + o : nullptr;
#pragma unroll
    for (int vo = 0; vo < JNT; ++vo) {
        const unsigned m = ADJ_MASK[vo];
        float a = 0.0f;
#pragma unroll
        for (int u = 0; u < JNT; ++u)
            if (m & (1u << u)) a += hn[u];
        a = fmaxf(a, 0.0f);
        if (rb) a += rb[vo * JNT];
        ob[vo * JNT] = a;
    }
}

// ---------------------------------------------------------------------------
// small per-vertex 17x17 conv + bias; accumulates BN sums via LDS reduction
__global__ __launch_bounds__(256) void small_conv(
    const float* __restrict__ F, const float* __restrict__ W,
    const float* __restrict__ bias, float* __restrict__ H,
    float* __restrict__ sums) {
    __shared__ float ls[2 * VSQ];
    for (int i = threadIdx.x; i < 2 * VSQ; i += 256) ls[i] = 0.0f;
    __syncthreads();

    int idx = blockIdx.x * blockDim.x + threadIdx.x;  // b*17 + v  (exact fit)
    if (idx < BATCH * JNT) {
        const int v = idx % JNT;
        const int b = idx / JNT;
        float fin[JNT];
        const float* fp = F + (size_t)b * VSQ + v * JNT;
#pragma unroll
        for (int d = 0; d < JNT; ++d) fin[d] = fp[d];
        const float* wv = W + v * VSQ;  // [d][o]
        float* hp = H + (size_t)b * VSQ + v * JNT;
#pragma unroll
        for (int o = 0; o < JNT; ++o) {
            float a = bias[v * JNT + o];
#pragma unroll
            for (int d = 0; d < JNT; ++d) a += fin[d] * wv[d * JNT + o];
            hp[o] = a;
            atomicAdd(&ls[v * JNT + o], a);
            atomicAdd(&ls[VSQ + v * JNT + o], a * a);
        }
    }
    __syncthreads();
    for (int i = threadIdx.x; i < 2 * VSQ; i += 256) atomicAdd(&sums[i], ls[i]);
}

// ---------------------------------------------------------------------------
extern "C" void kernel_launch(void* const* d_in, const int* in_sizes, int n_in,
                              void* d_out, int out_size, void* d_ws, size_t ws_size,
                              hipStream_t stream) {
    const float* img = (const float*)d_in[0];
    const float* W0  = (const float*)d_in[1];
    const float* b0  = (const float*)d_in[2];
    const float* g0  = (const float*)d_in[3];
    const float* be0 = (const float*)d_in[4];
    const float* Wr  = (const float*)d_in[5];
    const float* br  = (const float*)d_in[6];
    const float* gr  = (const float*)d_in[7];
    const float* ber = (const float*)d_in[8];
    float* out = (float*)d_out;

    float* ws    = (float*)d_ws;
    float* WT    = ws;                    // 17*32*2048
    float* F     = WT + JNT * NP * DIN;   // [B,17,17]
    float* G     = F + (size_t)BATCH * VSQ;
    float* H     = G + (size_t)BATCH * VSQ;
    float* SUMS  = H + (size_t)BATCH * VSQ;  // NBLK * 578
    float* STATS = SUMS + NBLK * 2 * VSQ;    // NBLK * 578

    zero_buf<<<(NBLK * 2 * VSQ + 255) / 256, 256, 0, stream>>>(SUMS, NBLK * 2 * VSQ);
    transpose_w0<<<(JNT * NP * DIN + 255) / 256, 256, 0, stream>>>(W0, WT);

    // block 0: 2048 -> 17 per-vertex GEMM (TDM + WMMA) + BN + adjacency + ReLU
    gemm_big<<<dim3(128, 17), 128, 0, stream>>>(img, WT, b0, H, SUMS);
    finalize_stats<<<2, 256, 0, stream>>>(SUMS, g0, be0, STATS);
    bn_adj_relu<<<(BATCH * JNT) / 256, 256, 0, stream>>>(H, STATS, nullptr, F);

    // 4 residual blocks x 2 conv blocks (17 -> 17)
    for (int r = 0; r < 4; ++r) {
        const int i0 = 2 * r, i1 = 2 * r + 1;
        float* s0p = SUMS + (size_t)(1 + i0) * 2 * VSQ;
        float* t0p = STATS + (size_t)(1 + i0) * 2 * VSQ;
        float* s1p = SUMS + (size_t)(1 + i1) * 2 * VSQ;
        float* t1p = STATS + (size_t)(1 + i1) * 2 * VSQ;

        small_conv<<<(BATCH * JNT) / 256, 256, 0, stream>>>(
            F, Wr + (size_t)i0 * JNT * VSQ, br + (size_t)i0 * VSQ, H, s0p);
        finalize_stats<<<2, 256, 0, stream>>>(s0p, gr + (size_t)i0 * VSQ,
                                              ber + (size_t)i0 * VSQ, t0p);
        bn_adj_relu<<<(BATCH * JNT) / 256, 256, 0, stream>>>(H, t0p, nullptr, G);

        small_conv<<<(BATCH * JNT) / 256, 256, 0, stream>>>(
            G, Wr + (size_t)i1 * JNT * VSQ, br + (size_t)i1 * VSQ, H, s1p);
        finalize_stats<<<2, 256, 0, stream>>>(s1p, gr + (size_t)i1 * VSQ,
                                              ber + (size_t)i1 * VSQ, t1p);
        // residual add (post-ReLU); last block writes the real output
        bn_adj_relu<<<(BATCH * JNT) / 256, 256, 0, stream>>>(
            H, t1p, F, (r == 3) ? out : F);
    }
}